// HeteroGNN_GraphLevel_27367531610690
// MI455X (gfx1250) — compile-verified
//
#include <hip/hip_runtime.h>
#include <hip/hip_bf16.h>

typedef float v2f __attribute__((ext_vector_type(2)));
typedef float v8f __attribute__((ext_vector_type(8)));

#define HDIM 128
#define GF_ACCUM 1
#define GF_RELU  2

static __device__ __forceinline__ float leaky02(float x) { return x > 0.f ? x : 0.2f * x; }

// order-preserving float<->uint encoding so segment-max maps to GLOBAL_ATOMIC_MAX_U32
static __device__ __forceinline__ unsigned encf(float x) {
    unsigned u = __float_as_uint(x);
    return (u & 0x80000000u) ? ~u : (u | 0x80000000u);
}
static __device__ __forceinline__ float decf(unsigned v) {
    return __uint_as_float((v & 0x80000000u) ? (v & 0x7fffffffu) : ~v);
}
static __device__ __forceinline__ float wave_sum(float v) {
    for (int off = 16; off; off >>= 1) v += __shfl_xor(v, off, 32);
    return v;
}

// ---------------- generic zero ----------------
__global__ void k_zero(float* p, long long n) {
    long long i = (long long)blockIdx.x * blockDim.x + threadIdx.x;
    long long stride = (long long)gridDim.x * blockDim.x;
    for (; i < n; i += stride) p[i] = 0.f;
}

// ---------------- WMMA GEMM: C[M x 128] (= / +=) X[M x K] @ W[K x 128] (+bias)(+relu) ----
// 8 waves/block; wave w computes the 16x16 tile at cols [16w,16w+16) of the block's 16-row stripe.
__global__ void k_gemm_wmma(const float* __restrict__ X, const float* __restrict__ W,
                            const float* __restrict__ bias, float* __restrict__ C,
                            int M, int K, int flags) {
    __shared__ float sA[16 * 256];
    const int wave = threadIdx.x >> 5;
    const int lane = threadIdx.x & 31;
    const int half = lane >> 4;   // 0: K pair {0,1}, 1: K pair {2,3} per WMMA step
    const int l16  = lane & 15;
    const int m0 = blockIdx.x * 16;

    const int tot = 16 * K;
    for (int i = threadIdx.x; i < tot; i += blockDim.x) {
        int r = i / K, c = i - r * K;
        int row = m0 + r; if (row >= M) row = M - 1;
        sA[i] = X[(size_t)row * K + c];
    }
    __syncthreads();

    const int n0 = wave * 16;
    const float bcol = bias ? bias[n0 + l16] : 0.f;
    v8f acc;
#pragma unroll
    for (int r = 0; r < 8; ++r) {
        int row = m0 + r + half * 8;
        float c0 = ((flags & GF_ACCUM) && row < M) ? C[(size_t)row * HDIM + n0 + l16] : 0.f;
        acc[r] = c0 + bcol;
    }
    for (int k = 0; k < K; k += 4) {
        const int kk = k + 2 * half;
        v2f a, b;
        a[0] = sA[l16 * K + kk];
        a[1] = sA[l16 * K + kk + 1];
        b[0] = W[(size_t)kk * HDIM + n0 + l16];
        b[1] = W[(size_t)(kk + 1) * HDIM + n0 + l16];
        acc = __builtin_amdgcn_wmma_f32_16x16x4_f32(false, a, false, b, (short)0, acc,
                                                    false, false);
    }
#pragma unroll
    for (int r = 0; r < 8; ++r) {
        int row = m0 + r + half * 8;
        if (row >= M) continue;
        float v = acc[r];
        if (flags & GF_RELU) v = fmaxf(v, 0.f);
        C[(size_t)row * HDIM + n0 + l16] = v;
    }
}

// ---------------- GAT helpers ----------------
// v[k] = sum_h W[k,h]*att[h]   (collapses the W_dst/W_src GEMMs used only for attention scores)
__global__ void k_watt(const float* __restrict__ W, const float* __restrict__ att, float* v) {
    int k = threadIdx.x;
    float s = 0.f;
    for (int h = 0; h < HDIM; ++h) s += W[(size_t)k * HDIM + h] * att[h];
    v[k] = s;
}

// per node (1 wave): a_s, a_d, and m initialized to the self-loop score
__global__ void k_node_scores(const float* __restrict__ X, const float* __restrict__ vs,
                              const float* __restrict__ vd, float* a_s, float* a_d,
                              unsigned* m_enc, int N) {
    int wave = threadIdx.x >> 5, lane = threadIdx.x & 31;
    int node = blockIdx.x * 8 + wave;
    if (node >= N) return;
    const float4 x = ((const float4*)(X + (size_t)node * HDIM))[lane];
    const float4 s4 = ((const float4*)vs)[lane];
    const float4 d4 = ((const float4*)vd)[lane];
    float ps = x.x * s4.x + x.y * s4.y + x.z * s4.z + x.w * s4.w;
    float pd = x.x * d4.x + x.y * d4.y + x.z * d4.z + x.w * d4.w;
    ps = wave_sum(ps);
    pd = wave_sum(pd);
    if (lane == 0) {
        a_s[node] = ps;
        a_d[node] = pd;
        m_enc[node] = encf(leaky02(ps + pd));
    }
}

__global__ void k_edge_max(const int* __restrict__ e, int E, const float* __restrict__ a_s,
                           const float* __restrict__ a_d, unsigned* m) {
    int i = blockIdx.x * blockDim.x + threadIdx.x;
    if (i >= E) return;
    int s = e[i], d = e[E + i];
    if (s == d) return;  // self-loops removed (re-added as fresh loops in init)
    atomicMax(&m[d], encf(leaky02(a_s[s] + a_d[d])));
}

// den initialized with the self-loop softmax weight (full write, no atomic needed)
__global__ void k_self_den(const float* __restrict__ a_s, const float* __restrict__ a_d,
                           const unsigned* __restrict__ m, float* den, int N) {
    int i = blockIdx.x * blockDim.x + threadIdx.x;
    if (i >= N) return;
    den[i] = __expf(leaky02(a_s[i] + a_d[i]) - decf(m[i]));
}

__global__ void k_edge_den(const int* __restrict__ e, int E, const float* __restrict__ a_s,
                           const float* __restrict__ a_d, const unsigned* __restrict__ m,
                           float* den) {
    int i = blockIdx.x * blockDim.x + threadIdx.x;
    if (i >= E) return;
    int s = e[i], d = e[E + i];
    if (s == d) return;
    atomicAdd(&den[d], __expf(leaky02(a_s[s] + a_d[d]) - decf(m[d])));
}

// Y[i,:] = alpha_self * xs[i,:] + bias   (full write; edges then atomic-accumulate on top)
__global__ void k_self_out(const float* __restrict__ xs, const float* __restrict__ a_s,
                           const float* __restrict__ a_d, const unsigned* __restrict__ m,
                           const float* __restrict__ den, const float* __restrict__ bias,
                           float* __restrict__ Y, int N) {
    int wave = threadIdx.x >> 5, lane = threadIdx.x & 31;
    int node = blockIdx.x * 8 + wave;
    if (node >= N) return;
    float alpha = __expf(leaky02(a_s[node] + a_d[node]) - decf(m[node])) / den[node];
    float4 x = ((const float4*)(xs + (size_t)node * HDIM))[lane];
    float4 b = ((const float4*)bias)[lane];
    float4 o = make_float4(alpha * x.x + b.x, alpha * x.y + b.y,
                           alpha * x.z + b.z, alpha * x.w + b.w);
    ((float4*)(Y + (size_t)node * HDIM))[lane] = o;
}

// one wave per edge: Y[d,:] += alpha * xs[s,:]  (coalesced 512B row per wave)
__global__ void k_edge_scatter(const int* __restrict__ e, int E, const float* __restrict__ xs,
                               const float* __restrict__ a_s, const float* __restrict__ a_d,
                               const unsigned* __restrict__ m, const float* __restrict__ den,
                               float* __restrict__ Y) {
    int wave = threadIdx.x >> 5, lane = threadIdx.x & 31;
    int ei = blockIdx.x * 8 + wave;
    if (ei >= E) return;
    int s = e[ei], d = e[E + ei];
    if (s == d) return;
    float alpha = __expf(leaky02(a_s[s] + a_d[d]) - decf(m[d])) / den[d];
    const float* xp = xs + (size_t)s * HDIM + lane * 4;
    float* yp = Y + (size_t)d * HDIM + lane * 4;
#pragma unroll
    for (int t = 0; t < 4; ++t) atomicAdd(&yp[t], alpha * xp[t]);
}

// ---------------- SAGE mean aggregation ----------------
__global__ void k_edge_agg(const int* __restrict__ e, int E, const float* __restrict__ Xsrc,
                           float* __restrict__ agg, float* __restrict__ cnt) {
    int wave = threadIdx.x >> 5, lane = threadIdx.x & 31;
    int ei = blockIdx.x * 8 + wave;
    if (ei >= E) return;
    int s = e[ei], d = e[E + ei];
    const float* xp = Xsrc + (size_t)s * HDIM + lane * 4;
    float* ap = agg + (size_t)d * HDIM + lane * 4;
#pragma unroll
    for (int t = 0; t < 4; ++t) atomicAdd(&ap[t], xp[t]);
    if (lane == 0) atomicAdd(&cnt[d], 1.f);
}

__global__ void k_mean(float* __restrict__ agg, const float* __restrict__ cnt, int N) {
    int wave = threadIdx.x >> 5, lane = threadIdx.x & 31;
    int node = blockIdx.x * 8 + wave;
    if (node >= N) return;
    float inv = 1.f / fmaxf(cnt[node], 1.f);
    float4* p = (float4*)(agg + (size_t)node * HDIM);
    float4 v = p[lane];
    v.x *= inv; v.y *= inv; v.z *= inv; v.w *= inv;
    p[lane] = v;
}

// ---------------- pooling + MLP tail ----------------
__global__ void k_pool(const float* __restrict__ h, const int* __restrict__ batch,
                       float* __restrict__ gemb, float* __restrict__ cntg,
                       int colOff, int cntOff, int N) {
    int wave = threadIdx.x >> 5, lane = threadIdx.x & 31;
    int node = blockIdx.x * 8 + wave;
    if (node >= N) return;
    int g = batch[node];
    const float* hp = h + (size_t)node * HDIM + lane * 4;
    float* gp = gemb + (size_t)g * 256 + colOff + lane * 4;
#pragma unroll
    for (int t = 0; t < 4; ++t) atomicAdd(&gp[t], hp[t]);
    if (lane == 0) atomicAdd(&cntg[cntOff + g], 1.f);
}

__global__ void k_pool_div(float* __restrict__ gemb, const float* __restrict__ cntg, int Gn) {
    int i = blockIdx.x * blockDim.x + threadIdx.x;
    if (i >= Gn * 256) return;
    int g = i >> 8, j = i & 255;
    float c = cntg[(j >= HDIM ? Gn : 0) + g];
    gemb[i] /= fmaxf(c, 1.f);
}

__global__ void k_final(const float* __restrict__ h1, const float* __restrict__ W2,
                        const float* __restrict__ b2, float* __restrict__ out, int Gn) {
    int wave = threadIdx.x >> 5, lane = threadIdx.x & 31;
    int g = blockIdx.x * 8 + wave;
    if (g >= Gn) return;
    float4 h = ((const float4*)(h1 + (size_t)g * HDIM))[lane];
    float4 w = ((const float4*)W2)[lane];
    float p = h.x * w.x + h.y * w.y + h.z * w.z + h.w * w.w;
    p = wave_sum(p);
    if (lane == 0) out[g] = p + b2[0];
}

// ---------------- orchestration ----------------
extern "C" void kernel_launch(void* const* d_in, const int* in_sizes, int n_in,
                              void* d_out, int out_size, void* d_ws, size_t ws_size,
                              hipStream_t stream) {
    auto cdiv = [](long long a, long long b) { return (int)((a + b - 1) / b); };

    const float* x_a = (const float*)d_in[0];
    const float* x_b = (const float*)d_in[1];
    const int* e_aa = (const int*)d_in[2];
    const int* e_ab = (const int*)d_in[3];
    const int* e_ba = (const int*)d_in[4];
    const int* batch_a = (const int*)d_in[5];
    const int* batch_b = (const int*)d_in[6];

    const int NAn = in_sizes[0] / HDIM;
    const int NBn = in_sizes[1] / HDIM;
    const int Eaa = in_sizes[2] / 2;
    const int Eab = in_sizes[3] / 2;
    const int Eba = in_sizes[4] / 2;
    const int Gn  = out_size;

    // params flattened in setup_inputs() dict insertion order
    struct Lp { const float *Wsrc, *Wdst, *atts, *attd, *baa,
                            *abWl, *abbl, *abWr, *baWl, *babl, *baWr; };
    Lp lp[2];
    for (int l = 0; l < 2; ++l) {
        int b = 7 + l * 11;
        lp[l].Wsrc = (const float*)d_in[b + 0];
        lp[l].Wdst = (const float*)d_in[b + 1];
        lp[l].atts = (const float*)d_in[b + 2];
        lp[l].attd = (const float*)d_in[b + 3];
        lp[l].baa  = (const float*)d_in[b + 4];
        lp[l].abWl = (const float*)d_in[b + 5];
        lp[l].abbl = (const float*)d_in[b + 6];
        lp[l].abWr = (const float*)d_in[b + 7];
        lp[l].baWl = (const float*)d_in[b + 8];
        lp[l].babl = (const float*)d_in[b + 9];
        lp[l].baWr = (const float*)d_in[b + 10];
    }
    const float* W1 = (const float*)d_in[29];
    const float* b1 = (const float*)d_in[30];
    const float* W2 = (const float*)d_in[31];
    const float* b2 = (const float*)d_in[32];
    float* out = (float*)d_out;

    // workspace layout
    float* ws = (float*)d_ws;
    const size_t NF = (size_t)NAn * HDIM;   // NA == NB in this problem
    float* haA = ws;            // layer0 output A
    float* hbA = ws + 1 * NF;   // layer0 output B
    float* haB = ws + 2 * NF;   // layer1 output A
    float* hbB = ws + 3 * NF;   // layer1 output B
    float* xs  = ws + 4 * NF;   // GAT projected source features
    float* agg = ws + 5 * NF;   // SAGE aggregation buffer
    float* sm  = ws + 6 * NF;
    float* a_s = sm;                    sm += NAn;
    float* a_d = sm;                    sm += NAn;
    unsigned* m = (unsigned*)sm;        sm += NAn;
    float* den = sm;                    sm += NAn;
    float* cnt = sm;                    sm += (NAn > NBn ? NAn : NBn);
    float* vs = sm;                     sm += HDIM;
    float* vd = sm;                     sm += HDIM;
    float* gemb = sm;                   sm += (size_t)Gn * 256;
    float* cntg = sm;                   sm += 2 * (size_t)Gn;
    float* h1 = sm;

    const int nbNodeA = cdiv(NAn, 8), nbNodeB = cdiv(NBn, 8);

    for (int l = 0; l < 2; ++l) {
        const float* Xa = l ? (const float*)haA : x_a;
        const float* Xb = l ? (const float*)hbA : x_b;
        float* Ya = l ? haB : haA;
        float* Yb = l ? hbB : hbA;
        const Lp& P = lp[l];

        // ---- GAT on A (edge_aa) ----
        k_watt<<<1, HDIM, 0, stream>>>(P.Wsrc, P.atts, vs);
        k_watt<<<1, HDIM, 0, stream>>>(P.Wdst, P.attd, vd);
        k_node_scores<<<nbNodeA, 256, 0, stream>>>(Xa, vs, vd, a_s, a_d, m, NAn);
        k_edge_max<<<cdiv(Eaa, 256), 256, 0, stream>>>(e_aa, Eaa, a_s, a_d, m);
        k_self_den<<<cdiv(NAn, 256), 256, 0, stream>>>(a_s, a_d, m, den, NAn);
        k_edge_den<<<cdiv(Eaa, 256), 256, 0, stream>>>(e_aa, Eaa, a_s, a_d, m, den);
        k_gemm_wmma<<<NAn / 16, 256, 0, stream>>>(Xa, P.Wsrc, nullptr, xs, NAn, HDIM, 0);
        k_self_out<<<nbNodeA, 256, 0, stream>>>(xs, a_s, a_d, m, den, P.baa, Ya, NAn);
        k_edge_scatter<<<cdiv(Eaa, 8), 256, 0, stream>>>(e_aa, Eaa, xs, a_s, a_d, m, den, Ya);

        // ---- SAGE b->a (edge_ba), fused into Ya ----
        k_zero<<<4096, 256, 0, stream>>>(agg, (long long)NAn * HDIM);
        k_zero<<<256, 256, 0, stream>>>(cnt, NAn);
        k_edge_agg<<<cdiv(Eba, 8), 256, 0, stream>>>(e_ba, Eba, Xb, agg, cnt);
        k_mean<<<nbNodeA, 256, 0, stream>>>(agg, cnt, NAn);
        k_gemm_wmma<<<NAn / 16, 256, 0, stream>>>(Xa, P.baWr, nullptr, Ya, NAn, HDIM,
                                                  GF_ACCUM);
        k_gemm_wmma<<<NAn / 16, 256, 0, stream>>>(agg, P.baWl, P.babl, Ya, NAn, HDIM,
                                                  GF_ACCUM | GF_RELU);

        // ---- SAGE a->b (edge_ab) -> Yb ----
        k_zero<<<4096, 256, 0, stream>>>(agg, (long long)NBn * HDIM);
        k_zero<<<256, 256, 0, stream>>>(cnt, NBn);
        k_edge_agg<<<cdiv(Eab, 8), 256, 0, stream>>>(e_ab, Eab, Xa, agg, cnt);
        k_mean<<<nbNodeB, 256, 0, stream>>>(agg, cnt, NBn);
        k_gemm_wmma<<<NBn / 16, 256, 0, stream>>>(agg, P.abWl, P.abbl, Yb, NBn, HDIM, 0);
        k_gemm_wmma<<<NBn / 16, 256, 0, stream>>>(Xb, P.abWr, nullptr, Yb, NBn, HDIM,
                                                  GF_ACCUM | GF_RELU);
    }

    // ---- mean pooling into gemb[G x 256] ----
    k_zero<<<256, 256, 0, stream>>>(gemb, (long long)Gn * 256 + 2 * Gn);  // gemb + cntg
    k_pool<<<nbNodeA, 256, 0, stream>>>(haB, batch_a, gemb, cntg, 0, 0, NAn);
    k_pool<<<nbNodeB, 256, 0, stream>>>(hbB, batch_b, gemb, cntg, HDIM, Gn, NBn);
    k_pool_div<<<cdiv((long long)Gn * 256, 256), 256, 0, stream>>>(gemb, cntg, Gn);

    // ---- MLP head ----
    k_gemm_wmma<<<Gn / 16, 256, 0, stream>>>(gemb, W1, b1, h1, Gn, 256, GF_RELU);
    k_final<<<cdiv(Gn, 8), 256, 0, stream>>>(h1, W2, b2, out, Gn);
}